// CapsuleLayer_9277129359362
// MI455X (gfx1250) — compile-verified
//
#include <hip/hip_runtime.h>
#include <hip/hip_bf16.h>
#include <math.h>

// CapsuleLayer routing for MI455X (gfx1250, wave32).
// Stage 1: fp32 WMMA batched GEMM (v_wmma_f32_16x16x4_f32), memory-bound.
// Stage 2: 3 routing iterations, b-update fused into the weighted reduction
//          so inputs_hat is streamed exactly once per iteration.

typedef __attribute__((ext_vector_type(2))) float v2f;
typedef __attribute__((ext_vector_type(8))) float v8f;

#define B_SZ     32
#define IN_CAPS  1152
#define IN_DIM   64
#define NUM_CAPS 32
#define OUT_DIM  64
#define ICHUNK   32
#define NCHUNKS  (IN_CAPS / ICHUNK)   // 36

// ---------------------------------------------------------------------------
// Kernel 1: inputs_hat[b,i,c,q] = sum_p x[b,i,p] * W[i,c,p,q]
// One wave per (i,c): M=32 (batch, 2 tiles), N=64 (q, 4 tiles), K=64 (p, 16 steps)
// A 16x4 fp32 layout: lanes 0-15 hold M rows; VGPR pair = K {0,1} (lo half) /
// {2,3} (hi half) of each 4-wide K window. B 4x16 mirrors it across lanes.
// ---------------------------------------------------------------------------
__global__ __launch_bounds__(256)
void caps_gemm_wmma(const float* __restrict__ x,
                    const float* __restrict__ W,
                    float* __restrict__ ih)
{
    const int i    = blockIdx.x;                    // input capsule
    const int wave = threadIdx.x >> 5;
    const int c    = blockIdx.y * 8 + wave;         // output capsule
    const int lane = threadIdx.x & 31;
    const int lr   = lane & 15;                     // M row / N col within tile
    const int lh   = lane >> 4;                     // K half select

    // Preload A: x rows for two M(batch) tiles, all K=64.
    v2f a[2][16];
#pragma unroll
    for (int mt = 0; mt < 2; ++mt) {
        const float* xp = x + ((size_t)(lr + mt * 16) * IN_CAPS + i) * IN_DIM + 2 * lh;
#pragma unroll
        for (int kk = 0; kk < 16; ++kk)
            a[mt][kk] = *(const v2f*)(xp + 4 * kk);
    }

    v8f acc[2][4] = {};   // 2 M-tiles x 4 N-tiles of 16x16 fp32

    const float* wbase = W + ((size_t)i * NUM_CAPS + c) * (IN_DIM * OUT_DIM);

#pragma unroll
    for (int kk = 0; kk < 16; ++kk) {
        // rows p = 4*kk + 2*lh, 4*kk + 2*lh + 1 ; col q = nt*16 + lr
        const float* wp = wbase + (size_t)(4 * kk + 2 * lh) * OUT_DIM + lr;
        if (kk < 15)
            __builtin_prefetch(wp + 4 * OUT_DIM, 0, 1);   // global_prefetch_b8
#pragma unroll
        for (int nt = 0; nt < 4; ++nt) {
            v2f bv;
            bv.x = wp[nt * 16];
            bv.y = wp[OUT_DIM + nt * 16];
            acc[0][nt] = __builtin_amdgcn_wmma_f32_16x16x4_f32(
                false, a[0][kk], false, bv, (short)0, acc[0][nt], false, false);
            acc[1][nt] = __builtin_amdgcn_wmma_f32_16x16x4_f32(
                false, a[1][kk], false, bv, (short)0, acc[1][nt], false, false);
        }
    }

    // C/D 16x16 layout: VGPR v -> M = v (lanes 0-15) or v+8 (lanes 16-31)
#pragma unroll
    for (int mt = 0; mt < 2; ++mt) {
#pragma unroll
        for (int v = 0; v < 8; ++v) {
            const int brow = mt * 16 + v + 8 * lh;
            float* op = ih + (((size_t)brow * IN_CAPS + i) * NUM_CAPS + c) * OUT_DIM + lr;
#pragma unroll
            for (int nt = 0; nt < 4; ++nt)
                op[nt * 16] = acc[mt][nt][v];
        }
    }
}

// ---------------------------------------------------------------------------
// Kernel 2: fused routing step.
// mode 0 (r=0): logits = bias                      (no dot, no blog write)
// mode 1 (r=1): logits = bias + dot(ih, out_prev); blog := logits
// mode 2 (r=2): logits = blog + dot(ih, out_prev)  (no write needed)
// Each block owns (batch b, chunk of 32 i's); accumulates c*ih partials in
// registers and writes its own slice of spart (no fp atomics -> deterministic).
// ---------------------------------------------------------------------------
__global__ __launch_bounds__(256)
void caps_route(const float* __restrict__ ih,
                const float* __restrict__ bias,
                float* __restrict__ blog,
                const float* __restrict__ outprev,
                float* __restrict__ spart,
                int mode)
{
    __shared__ float lds_d[NUM_CAPS];
    __shared__ float lds_p[NUM_CAPS];

    const int b     = blockIdx.x;
    const int chunk = blockIdx.y;
    const int t     = threadIdx.x;
    const int q     = t & 63;

    float accum[8] = {0.f, 0.f, 0.f, 0.f, 0.f, 0.f, 0.f, 0.f};

    const int dc  = t >> 3;   // dot phase: c per group of 8 threads
    const int dq0 = t & 7;

    for (int ii = 0; ii < ICHUNK; ++ii) {
        const int i = chunk * ICHUNK + ii;
        const float* ihrow = ih + (((size_t)b * IN_CAPS + i) * NUM_CAPS) * OUT_DIM;

        if (mode != 0) {
            const float* ihc = ihrow + dc * OUT_DIM;
            const float* oc  = outprev + ((size_t)b * NUM_CAPS + dc) * OUT_DIM;
            float d = 0.f;
#pragma unroll
            for (int j = 0; j < 8; ++j) {
                const int qq = dq0 + 8 * j;
                d += ihc[qq] * oc[qq];
            }
            d += __shfl_xor(d, 1);   // 8-thread groups live within one wave32
            d += __shfl_xor(d, 2);
            d += __shfl_xor(d, 4);
            if (dq0 == 0) lds_d[dc] = d;
        }
        __syncthreads();

        if (t < NUM_CAPS) {   // wave 0 does the 32-wide softmax
            const size_t bi = ((size_t)b * IN_CAPS + i) * NUM_CAPS + t;
            float base = (mode == 2) ? blog[bi] : bias[(size_t)i * NUM_CAPS + t];
            float l = base + ((mode != 0) ? lds_d[t] : 0.f);
            if (mode == 1) blog[bi] = l;
            float m = l;
#pragma unroll
            for (int off = 16; off > 0; off >>= 1)
                m = fmaxf(m, __shfl_xor(m, off));
            float e = __expf(l - m);
            float s = e;
#pragma unroll
            for (int off = 16; off > 0; off >>= 1)
                s += __shfl_xor(s, off);
            lds_p[t] = e / s;
        }
        __syncthreads();

#pragma unroll
        for (int k = 0; k < 8; ++k) {
            const int cc = (t >> 6) + 4 * k;
            accum[k] += lds_p[cc] * ihrow[cc * OUT_DIM + q];
        }
        // next iteration's lds writes are fenced by its own __syncthreads()
    }

    float* sp = spart + (((size_t)chunk * B_SZ + b) * NUM_CAPS) * OUT_DIM;
#pragma unroll
    for (int k = 0; k < 8; ++k) {
        const int cc = (t >> 6) + 4 * k;
        sp[cc * OUT_DIM + q] = accum[k];
    }
}

// ---------------------------------------------------------------------------
// Kernel 3: reduce chunk partials and apply squash over q.
// Block = (b,c), 64 threads (one per q).
// ---------------------------------------------------------------------------
__global__ __launch_bounds__(64)
void caps_squash(const float* __restrict__ spart, float* __restrict__ out)
{
    __shared__ float lds[2];
    const int bc = blockIdx.x;       // b*32 + c
    const int q  = threadIdx.x;      // 0..63

    float v = 0.f;
    for (int ch = 0; ch < NCHUNKS; ++ch)
        v += spart[((size_t)ch * (B_SZ * NUM_CAPS) + bc) * OUT_DIM + q];

    float sq = v * v;
#pragma unroll
    for (int off = 16; off > 0; off >>= 1)
        sq += __shfl_xor(sq, off);
    if ((q & 31) == 0) lds[q >> 5] = sq;
    __syncthreads();
    const float s2    = lds[0] + lds[1];
    const float scale = s2 / (1.f + s2) * rsqrtf(s2);
    out[(size_t)bc * OUT_DIM + q] = scale * v;
}

// ---------------------------------------------------------------------------
extern "C" void kernel_launch(void* const* d_in, const int* in_sizes, int n_in,
                              void* d_out, int out_size, void* d_ws, size_t ws_size,
                              hipStream_t stream)
{
    const float* x    = (const float*)d_in[0];   // [32,1152,64]
    const float* W    = (const float*)d_in[1];   // [1152,32,64,64]
    const float* bias = (const float*)d_in[2];   // [1,1152,32,1]
    float* out = (float*)d_out;                  // [32,32,64]

    char* ws = (char*)d_ws;
    float* ih = (float*)ws;                                           // 302 MB
    const size_t ih_bytes = (size_t)B_SZ * IN_CAPS * NUM_CAPS * OUT_DIM * sizeof(float);
    float* blog = (float*)(ws + ih_bytes);                            // 4.7 MB
    const size_t blog_bytes = (size_t)B_SZ * IN_CAPS * NUM_CAPS * sizeof(float);
    float* spart = (float*)(ws + ih_bytes + blog_bytes);              // 9.4 MB
    const size_t spart_bytes = (size_t)NCHUNKS * B_SZ * NUM_CAPS * OUT_DIM * sizeof(float);
    float* outputs = (float*)(ws + ih_bytes + blog_bytes + spart_bytes); // 256 KB

    // Stage 1: inputs_hat via fp32 WMMA (1152 x 4 blocks, 8 waves each -> one
    // wave per (i,c) pair).
    caps_gemm_wmma<<<dim3(IN_CAPS, NUM_CAPS / 8), 256, 0, stream>>>(x, W, ih);

    // Stage 2: 3 routing iterations, inputs_hat streamed once each.
    for (int r = 0; r < 3; ++r) {
        caps_route<<<dim3(B_SZ, NCHUNKS), 256, 0, stream>>>(
            ih, bias, blog, outputs, spart, r);
        float* dst = (r == 2) ? out : outputs;
        caps_squash<<<dim3(B_SZ * NUM_CAPS), 64, 0, stream>>>(spart, dst);
    }
}